// EmbedderNeuronGroup_16904991277077
// MI455X (gfx1250) — compile-verified
//
#include <hip/hip_runtime.h>

typedef __attribute__((ext_vector_type(2))) float v2f;
typedef __attribute__((ext_vector_type(8))) float v8f;

#define BATCH   2048
#define DMODEL  1024
#define NG2     26
#define F2      80
#define NG1     24
#define F1      16
#define INDIM   (NG2 * F2 + NG1 * F1)   /* 2464 */
#define NGROUPS (NG2 + NG1)             /* 50   */

// One wave (32 lanes) computes a 16(batch) x 16(d_model) output tile for all
// 50 groups. f32 WMMA 16x16x4, K-chained (20 steps for the width-80 slices,
// 4 steps for the width-16 slices). B/W fragments preloaded per wave.
__global__ __launch_bounds__(128) void embedder_wmma_f32_kernel(
    const float* __restrict__ x,
    const float* __restrict__ W1, const float* __restrict__ b1,
    const float* __restrict__ W2, const float* __restrict__ b2,
    float* __restrict__ out)
{
    const int lane  = threadIdx.x & 31;
    const int wid   = threadIdx.x >> 5;
    const int n0    = blockIdx.x * 64 + wid * 16;   // d_model tile base
    const int m0    = blockIdx.y * 16;              // batch-row tile base

    const int nlane = lane & 15;         // N (column) / M (row) sub-index
    const int khalf = (lane >> 4) * 2;   // A/B fragment K offset: lanes 16-31 hold K+2
    const int mhalf = (lane >> 4) * 8;   // C/D: lanes 16-31 hold rows M+8

    // ---- Preload B fragments (weights, transposed view: B[k][n] = W[n][k]) ----
    v2f Bw2[20];
    {
        const float* wrow = W2 + (size_t)(n0 + nlane) * F2;
#pragma unroll
        for (int ks = 0; ks < 20; ++ks) {
            Bw2[ks].x = wrow[ks * 4 + khalf + 0];
            Bw2[ks].y = wrow[ks * 4 + khalf + 1];
        }
    }
    v2f Bw1[4];
    {
        const float* wrow = W1 + (size_t)(n0 + nlane) * F1;
#pragma unroll
        for (int ks = 0; ks < 4; ++ks) {
            Bw1[ks].x = wrow[ks * 4 + khalf + 0];
            Bw1[ks].y = wrow[ks * 4 + khalf + 1];
        }
    }
    const float bias2 = b2[n0 + nlane];
    const float bias1 = b1[n0 + nlane];

    // A fragments: lane holds batch row (m0 + nlane), K pair at khalf
    const float* xrow = x + (size_t)(m0 + nlane) * INDIM;

    for (int g = 0; g < NGROUPS; ++g) {   // uniform across wave -> EXEC all 1s
        v8f  c = {};
        float bias;
        if (g < NG2) {
            const float* a0 = xrow + g * F2 + khalf;
#pragma unroll
            for (int ks = 0; ks < 20; ++ks) {
                v2f a;
                a.x = a0[ks * 4 + 0];
                a.y = a0[ks * 4 + 1];
                c = __builtin_amdgcn_wmma_f32_16x16x4_f32(
                        /*neg_a=*/false, a, /*neg_b=*/false, Bw2[ks],
                        /*c_mod=*/(short)0, c, /*reuse_a=*/false, /*reuse_b=*/false);
            }
            bias = bias2;
        } else {
            const float* a0 = xrow + NG2 * F2 + (g - NG2) * F1 + khalf;
#pragma unroll
            for (int ks = 0; ks < 4; ++ks) {
                v2f a;
                a.x = a0[ks * 4 + 0];
                a.y = a0[ks * 4 + 1];
                c = __builtin_amdgcn_wmma_f32_16x16x4_f32(
                        false, a, false, Bw1[ks],
                        (short)0, c, false, false);
            }
            bias = bias1;
        }

        // Store 16x16 tile: vgpr r -> row (mhalf + r), col nlane.
        // Output is write-once streamed (419 MB) -> non-temporal stores to
        // keep L2 (192 MB) holding x and the weights instead.
        float* obase = out + ((size_t)(m0 + mhalf) * NGROUPS + g) * DMODEL + n0 + nlane;
#pragma unroll
        for (int r = 0; r < 8; ++r) {
            __builtin_nontemporal_store(c[r] + bias, obase + (size_t)r * NGROUPS * DMODEL);
        }
    }
}

extern "C" void kernel_launch(void* const* d_in, const int* in_sizes, int n_in,
                              void* d_out, int out_size, void* d_ws, size_t ws_size,
                              hipStream_t stream) {
    // setup_inputs() order: x, W1, b1, W2, b2  (all float32)
    const float* x  = (const float*)d_in[0];
    const float* W1 = (const float*)d_in[1];
    const float* b1 = (const float*)d_in[2];
    const float* W2 = (const float*)d_in[3];
    const float* b2 = (const float*)d_in[4];
    float* out = (float*)d_out;

    dim3 grid(DMODEL / 64, BATCH / 16);   // 16 x 128 blocks
    dim3 block(128);                      // 4 waves; each wave = one 16x16 N-tile
    embedder_wmma_f32_kernel<<<grid, block, 0, stream>>>(x, W1, b1, W2, b2, out);
}